// GCNNodeClassifier_9466107920639
// MI455X (gfx1250) — compile-verified
//
#include <hip/hip_runtime.h>

typedef __attribute__((ext_vector_type(2))) float v2f;
typedef __attribute__((ext_vector_type(8))) float v8f;

#define HDIM 128

// ---------------- utility kernels ----------------

__global__ __launch_bounds__(256) void zero_kernel(float* __restrict__ p, int n) {
  int i = blockIdx.x * 256 + threadIdx.x;
  if (i < n) p[i] = 0.0f;
}

// vectorized zero: n4 = number of float4 groups
__global__ __launch_bounds__(256) void zero4_kernel(float4* __restrict__ p, long long n4) {
  long long i = (long long)blockIdx.x * 256 + threadIdx.x;
  if (i < n4) p[i] = make_float4(0.f, 0.f, 0.f, 0.f);
}

// deg[d] += 1 for each edge destination (self-loop added in dinv_kernel)
__global__ __launch_bounds__(256) void deg_kernel(const long long* __restrict__ ei,
                                                  float* __restrict__ deg, int E) {
  int e = blockIdx.x * 256 + threadIdx.x;
  if (e < E) {
    long long d = ei[(long long)E + e];
    __hip_atomic_fetch_add(&deg[d], 1.0f, __ATOMIC_RELAXED, __HIP_MEMORY_SCOPE_AGENT);
  }
}

__global__ __launch_bounds__(256) void dinv_kernel(float* __restrict__ dinv, int n) {
  int i = blockIdx.x * 256 + threadIdx.x;
  if (i < n) dinv[i] = rsqrtf(dinv[i] + 1.0f);  // +1 = self loop; deg>=1 always
}

// -------- WMMA GEMM + row scale: P[N,128] = dinv ⊙ (A[N,128] @ B[128,128]) -----
// One wave per 16x16 tile; 8 waves/block cover the 8 column tiles of HDIM.
__global__ __launch_bounds__(256) void gemm_scale_kernel(const float* __restrict__ A,
                                                         const float* __restrict__ B,
                                                         const float* __restrict__ dinv,
                                                         float* __restrict__ P, int N) {
  const int lane = threadIdx.x & 31;
  const int wid  = threadIdx.x >> 5;   // column tile 0..7
  const int m    = lane & 15;
  const int kh   = lane >> 4;          // K-half select (0/1)

  int row = blockIdx.x * 16 + m;
  if (row >= N) row = N - 1;           // clamp (stores are guarded on slow path)
  const float* __restrict__ arow = A + (long long)row * HDIM;
  const int col = wid * 16 + m;

  v8f acc = {};
#pragma unroll 8
  for (int k0 = 0; k0 < HDIM; k0 += 4) {
    const int ka = k0 + 2 * kh;
    // A 16x4 fragment: lane m holds A[m][ka], A[m][ka+1]
    v2f av = *(const v2f*)(arow + ka);
    // B 4x16 fragment: lane holds B[ka][col], B[ka+1][col]
    v2f bv;
    bv.x = B[(ka + 0) * HDIM + col];
    bv.y = B[(ka + 1) * HDIM + col];
    acc = __builtin_amdgcn_wmma_f32_16x16x4_f32(
        /*neg_a=*/false, av, /*neg_b=*/false, bv,
        /*c_mod=*/(short)0, acc, /*reuse_a=*/false, /*reuse_b=*/false);
  }

  // C/D layout: VGPR r -> row (r + 8*kh), col = lane&15
  const int c = lane & 15;
  const int base = blockIdx.x * 16;
  if (base + 16 <= N) {
    // fast path: full tile, unguarded stores
#pragma unroll
    for (int r = 0; r < 8; ++r) {
      const int orow = base + r + 8 * kh;
      const float dv = dinv[orow];
      P[(long long)orow * HDIM + wid * 16 + c] = dv * acc[r];
    }
  } else {
#pragma unroll
    for (int r = 0; r < 8; ++r) {
      const int orow = base + r + 8 * kh;
      if (orow < N) {
        const float dv = dinv[orow];
        P[(long long)orow * HDIM + wid * 16 + c] = dv * acc[r];
      }
    }
  }
}

// ---------------- WMMA FC: OUT[N,C] = A[N,128] @ B[128,C] + bias ----------------
__global__ void fc_kernel(const float* __restrict__ A, const float* __restrict__ B,
                          const float* __restrict__ bias, float* __restrict__ OUT,
                          int N, int C) {
  const int lane = threadIdx.x & 31;
  const int wid  = threadIdx.x >> 5;   // column tile
  const int m    = lane & 15;
  const int kh   = lane >> 4;

  int row = blockIdx.x * 16 + m;
  if (row >= N) row = N - 1;
  const float* __restrict__ arow = A + (long long)row * HDIM;
  const int colg = wid * 16 + m;
  const bool colok = (colg < C);

  v8f acc = {};
#pragma unroll 4
  for (int k0 = 0; k0 < HDIM; k0 += 4) {
    const int ka = k0 + 2 * kh;
    v2f av = *(const v2f*)(arow + ka);
    v2f bv;
    bv.x = colok ? B[(long long)(ka + 0) * C + colg] : 0.0f;
    bv.y = colok ? B[(long long)(ka + 1) * C + colg] : 0.0f;
    acc = __builtin_amdgcn_wmma_f32_16x16x4_f32(
        false, av, false, bv, (short)0, acc, false, false);
  }

  const int c = lane & 15;
  const int colo = wid * 16 + c;
  const int base = blockIdx.x * 16;
  if (colo < C) {
    const float bb = bias[colo];
    if (base + 16 <= N) {
#pragma unroll
      for (int r = 0; r < 8; ++r) {
        const int orow = base + r + 8 * kh;
        OUT[(long long)orow * C + colo] = acc[r] + bb;
      }
    } else {
#pragma unroll
      for (int r = 0; r < 8; ++r) {
        const int orow = base + r + 8 * kh;
        if (orow < N) OUT[(long long)orow * C + colo] = acc[r] + bb;
      }
    }
  }
}

// ---------------- edge scatter: S[dst] += P[src], one wave per edge ------------
__global__ __launch_bounds__(256) void edge_agg_kernel(const long long* __restrict__ ei,
                                                       const float* __restrict__ P,
                                                       float* __restrict__ S, int E) {
  long long t = (long long)blockIdx.x * 256 + threadIdx.x;
  int e = (int)(t >> 5);
  int lane = (int)(t & 31);
  if (e >= E) return;
  long long s = ei[e];                 // src
  long long d = ei[(long long)E + e];  // dst
  const float4 v = *(const float4*)(P + s * HDIM + lane * 4);
  float* __restrict__ dst = S + d * HDIM + lane * 4;
  __hip_atomic_fetch_add(dst + 0, v.x, __ATOMIC_RELAXED, __HIP_MEMORY_SCOPE_AGENT);
  __hip_atomic_fetch_add(dst + 1, v.y, __ATOMIC_RELAXED, __HIP_MEMORY_SCOPE_AGENT);
  __hip_atomic_fetch_add(dst + 2, v.z, __ATOMIC_RELAXED, __HIP_MEMORY_SCOPE_AGENT);
  __hip_atomic_fetch_add(dst + 3, v.w, __ATOMIC_RELAXED, __HIP_MEMORY_SCOPE_AGENT);
}

// ------- epilogue: OUT = [relu]( dinv ⊙ (S + P) + b )   (P adds the self-loop) --
__global__ __launch_bounds__(256) void epilogue_kernel(const float* __restrict__ dinv,
                                                       const float* __restrict__ S,
                                                       const float* __restrict__ P,
                                                       const float* __restrict__ bias,
                                                       float* __restrict__ OUT,
                                                       int N, int relu) {
  long long t = (long long)blockIdx.x * 256 + threadIdx.x;
  long long total = (long long)N * (HDIM / 4);
  if (t >= total) return;
  long long node = t >> 5;
  int jg = (int)(t & 31) * 4;
  float dv = dinv[node];
  float4 s = *(const float4*)(S + t * 4);
  float4 p = *(const float4*)(P + t * 4);
  float4 b = *(const float4*)(bias + jg);
  float4 o;
  o.x = (s.x + p.x) * dv + b.x;
  o.y = (s.y + p.y) * dv + b.y;
  o.z = (s.z + p.z) * dv + b.z;
  o.w = (s.w + p.w) * dv + b.w;
  if (relu) {
    o.x = fmaxf(o.x, 0.0f); o.y = fmaxf(o.y, 0.0f);
    o.z = fmaxf(o.z, 0.0f); o.w = fmaxf(o.w, 0.0f);
  }
  *(float4*)(OUT + t * 4) = o;
}

// ---------------- driver -------------------------------------------------------

extern "C" void kernel_launch(void* const* d_in, const int* in_sizes, int n_in,
                              void* d_out, int out_size, void* d_ws, size_t ws_size,
                              hipStream_t stream) {
  const float*     x   = (const float*)d_in[0];
  const long long* ei  = (const long long*)d_in[1];
  const float*     W1  = (const float*)d_in[2];
  const float*     b1  = (const float*)d_in[3];
  const float*     W2  = (const float*)d_in[4];
  const float*     b2  = (const float*)d_in[5];
  const float*     Wfc = (const float*)d_in[6];
  const float*     bfc = (const float*)d_in[7];
  float* out = (float*)d_out;

  const int N = in_sizes[0] / HDIM;   // 50000
  const int E = in_sizes[1] / 2;      // 600000
  const int C = in_sizes[7];          // 40

  // workspace layout: dinv[N] | tmp1[N*128] | tmp2[N*128]
  float* dinv = (float*)d_ws;
  float* tmp1 = dinv + (size_t)((N + 255) / 256) * 256;
  float* tmp2 = tmp1 + (size_t)N * HDIM;

  const int mtiles = (N + 15) / 16;
  const long long ngroups = (long long)N * (HDIM / 4);     // float4 groups per feature buf
  const int gblocks = (int)((ngroups + 255) / 256);
  const long long ethreads = (long long)E * 32;
  const int eblocks = (int)((ethreads + 255) / 256);

  // degrees -> dinv
  zero_kernel<<<(N + 255) / 256, 256, 0, stream>>>(dinv, N);
  deg_kernel<<<(E + 255) / 256, 256, 0, stream>>>(ei, dinv, E);
  dinv_kernel<<<(N + 255) / 256, 256, 0, stream>>>(dinv, N);

  // ----- layer 1: h1 = relu(dinv ⊙ (scatter(P) + P) + b1), P = dinv ⊙ (x·W1) ---
  gemm_scale_kernel<<<mtiles, 256, 0, stream>>>(x, W1, dinv, tmp1, N);     // P -> tmp1
  zero4_kernel<<<gblocks, 256, 0, stream>>>((float4*)tmp2, ngroups);       // S -> 0
  edge_agg_kernel<<<eblocks, 256, 0, stream>>>(ei, tmp1, tmp2, E);         // S += P[src]
  epilogue_kernel<<<gblocks, 256, 0, stream>>>(dinv, tmp2, tmp1, b1, tmp1, N, 1);  // h1->tmp1

  // ----- layer 2 (no relu) -----
  gemm_scale_kernel<<<mtiles, 256, 0, stream>>>(tmp1, W2, dinv, tmp2, N);  // P -> tmp2
  zero4_kernel<<<gblocks, 256, 0, stream>>>((float4*)tmp1, ngroups);       // S -> 0
  edge_agg_kernel<<<eblocks, 256, 0, stream>>>(ei, tmp2, tmp1, E);         // S += P[src]
  epilogue_kernel<<<gblocks, 256, 0, stream>>>(dinv, tmp1, tmp2, b2, tmp2, N, 0);  // h2->tmp2

  // ----- FC head: out = h2 @ Wfc + bfc -----
  const int ntilesC = (C + 15) / 16;
  fc_kernel<<<mtiles, 32 * ntilesC, 0, stream>>>(tmp2, Wfc, bfc, out, N, C);
}